// CorVolume_34437047779562
// MI455X (gfx1250) — compile-verified
//
#include <hip/hip_runtime.h>

// CDNA5 / gfx1250 stereo correlation cost volume via V_WMMA_F32_16X16X4_F32.
// out[b,d,h,w] = (1/64) * sum_c left[b,c,h,w] * right[b,c,h,w-d], 0 if w<d.
// B=8 C=64 H=128 W=256 D=64. One workgroup per (b,h); banded GEMM in LDS.
// Staging uses CDNA5 async global->LDS copies (ASYNCcnt) when the builtin is
// available; output uses non-temporal stores (streamed once, never re-read).

typedef __attribute__((ext_vector_type(2))) float v2f;
typedef __attribute__((ext_vector_type(4))) float v4f;
typedef __attribute__((ext_vector_type(8))) float v8f;
typedef __attribute__((ext_vector_type(4))) int   v4i;

#define B_DIM 8
#define C_DIM 64
#define H_DIM 128
#define W_DIM 256
#define D_DIM 64
// Padded LDS row stride (floats): 260*4 = 1040 bytes = 65*16 -> 16B aligned rows,
// and rows c, c+2 land on different bank groups (520 mod 64 != 0): no 2-way
// conflict between lane halves when reading A/B fragments.
#define LDS_STRIDE 260

#if __has_builtin(__builtin_amdgcn_global_load_async_to_lds_b128)
#define USE_ASYNC_LDS 1
// Builtin wants int4 pointers: global source in AS(1), LDS dest in AS(3).
typedef __attribute__((address_space(1))) v4i gv4i;
typedef __attribute__((address_space(3))) v4i lv4i;
#else
#define USE_ASYNC_LDS 0
#endif

__global__ __launch_bounds__(512)
void corr_volume_wmma_kernel(const float* __restrict__ left,
                             const float* __restrict__ right,
                             float* __restrict__ out)
{
    __shared__ float smem[2 * C_DIM * LDS_STRIDE];  // ~130 KB of the 320 KB WGP LDS
    float* lds_l = smem;
    float* lds_r = smem + C_DIM * LDS_STRIDE;

    const int bh = blockIdx.x;        // 0 .. B*H-1
    const int b  = bh / H_DIM;
    const int h  = bh % H_DIM;

    const size_t chan_stride = (size_t)H_DIM * W_DIM;           // 32768 floats
    const float* gl = left  + (size_t)b * C_DIM * chan_stride + (size_t)h * W_DIM;
    const float* gr = right + (size_t)b * C_DIM * chan_stride + (size_t)h * W_DIM;

    // ---- Stage both rows (64 channels x 256 w, fp32) into LDS, c-major ----
    const int tid = threadIdx.x;      // 0..511
#if USE_ASYNC_LDS
    {
        #pragma unroll
        for (int i = tid; i < C_DIM * (W_DIM / 4); i += 512) {  // 4096 b128 copies/matrix
            const int c  = i >> 6;        // / 64
            const int w4 = (i & 63) << 2; // * 4
            __builtin_amdgcn_global_load_async_to_lds_b128(
                (gv4i*)(gl + (size_t)c * chan_stride + w4),
                (lv4i*)(lds_l + c * LDS_STRIDE + w4), 0, 0);
            __builtin_amdgcn_global_load_async_to_lds_b128(
                (gv4i*)(gr + (size_t)c * chan_stride + w4),
                (lv4i*)(lds_r + c * LDS_STRIDE + w4), 0, 0);
        }
#if __has_builtin(__builtin_amdgcn_s_wait_asynccnt)
        __builtin_amdgcn_s_wait_asynccnt(0);
#else
        asm volatile("s_wait_asynccnt 0" ::: "memory");
#endif
    }
#else
    #pragma unroll
    for (int i = tid; i < C_DIM * (W_DIM / 4); i += 512) {      // 4096 float4 chunks
        const int c  = i >> 6;        // / 64
        const int w4 = (i & 63) << 2; // * 4
        v4f vl = *(const v4f*)(gl + (size_t)c * chan_stride + w4);
        v4f vr = *(const v4f*)(gr + (size_t)c * chan_stride + w4);
        *(v4f*)(lds_l + c * LDS_STRIDE + w4) = vl;
        *(v4f*)(lds_r + c * LDS_STRIDE + w4) = vr;
    }
#endif
    __syncthreads();

    // ---- Per-wave banded GEMM: wave i owns w-tile [16i, 16i+16) ----
    const int lane = tid & 31;
    const int wave = tid >> 5;            // 0..15
    const int w0   = wave << 4;
    const int n    = lane & 15;           // A: row M; B: col N; store: col N
    const int kh   = (lane >> 4) << 1;    // K sub-offset 0 or 2 per lane half
    const int mh   = (lane >> 4) << 3;    // C/D: M base 0 or 8 per lane half

    // Load full-K A fragment once: A[M=w0+n][K], K = 4*kc + kh + v
    float a[32];
    #pragma unroll
    for (int kc = 0; kc < 16; ++kc) {
        a[2 * kc + 0] = lds_l[(kc * 4 + kh + 0) * LDS_STRIDE + w0 + n];
        a[2 * kc + 1] = lds_l[(kc * 4 + kh + 1) * LDS_STRIDE + w0 + n];
    }

    const float scale = 1.0f / (float)C_DIM;
    float* outb = out + (size_t)b * D_DIM * chan_stride + (size_t)h * W_DIM;

    // Five w'-tiles intersect the 64-wide band below the diagonal.
    #pragma unroll
    for (int oi = 0; oi < 5; ++oi) {
        const int o   = oi << 4;
        const int w0p = __builtin_amdgcn_readfirstlane(w0 - o);  // wave-uniform scalar

        v8f acc = {0.f, 0.f, 0.f, 0.f, 0.f, 0.f, 0.f, 0.f};
        if (w0p >= 0) {   // scalar branch: EXEC stays all-1s around WMMA
            #pragma unroll
            for (int kc = 0; kc < 16; ++kc) {
                v2f av = { a[2 * kc + 0], a[2 * kc + 1] };
                v2f bv = { lds_r[(kc * 4 + kh + 0) * LDS_STRIDE + w0p + n],
                           lds_r[(kc * 4 + kh + 1) * LDS_STRIDE + w0p + n] };
                acc = __builtin_amdgcn_wmma_f32_16x16x4_f32(
                        /*neg_a=*/false, av, /*neg_b=*/false, bv,
                        /*c_mod=*/(short)0, acc,
                        /*reuse_a=*/false, /*reuse_b=*/false);
            }
        }
        // Store D tile: lane holds (M = j+mh, N = n); d = w - w' = o + M - n.
        // Band predicate writes each output element exactly once; out-of-range
        // w' tiles (w0p<0) store the zero pad here, so no separate fill pass.
        // Non-temporal: output is streamed once, never re-read by this kernel.
        #pragma unroll
        for (int j = 0; j < 8; ++j) {
            const int M = j + mh;
            const int d = o + M - n;
            if (0 <= d && d < D_DIM) {
                __builtin_nontemporal_store(acc[j] * scale,
                    outb + (size_t)d * chan_stride + (w0 + M));
            }
        }
    }
}

extern "C" void kernel_launch(void* const* d_in, const int* in_sizes, int n_in,
                              void* d_out, int out_size, void* d_ws, size_t ws_size,
                              hipStream_t stream) {
    (void)in_sizes; (void)n_in; (void)d_ws; (void)ws_size; (void)out_size;
    const float* left  = (const float*)d_in[0];
    const float* right = (const float*)d_in[1];
    // d_in[2] is max_disp (== 64), compile-time constant here.
    float* out = (float*)d_out;

    dim3 grid(B_DIM * H_DIM);   // 1024 workgroups, one per (b,h) row
    dim3 block(512);            // 16 wave32 waves, one per 16-wide w tile
    corr_volume_wmma_kernel<<<grid, block, 0, stream>>>(left, right, out);
}